// GeometricGraphTransformer_83270825935561
// MI455X (gfx1250) — compile-verified
//
#include <hip/hip_runtime.h>
#include <math.h>

// ---------------- problem constants (from reference) ----------------
#define NN 50000
#define EE 600000
#define DD 128
#define HH 8
#define DHH 16
#define EDIM 16
#define NLAYER 2
#define LN_EPS 1e-5f

typedef __attribute__((ext_vector_type(2))) float v2f;
typedef __attribute__((ext_vector_type(8))) float v8f;

// ---------------- small helpers ----------------
__device__ __forceinline__ void atomicMaxF(float* addr, float val) {
    unsigned int* ua = (unsigned int*)addr;
    unsigned int old = *ua;
    while (__uint_as_float(old) < val) {
        unsigned int assumed = old;
        old = atomicCAS(ua, assumed, __float_as_uint(val));
        if (old == assumed) break;
    }
}

__global__ void __launch_bounds__(256)
fill_kernel(float* __restrict__ p, float v, int n) {
    int i = blockIdx.x * blockDim.x + threadIdx.x;
    if (i < n) p[i] = v;
}

// ---------------- LayerNorm (optionally fused residual add) ----------------
// one wave32 per row of 128; 4 floats per lane; shuffle reduction
__global__ void __launch_bounds__(256)
ln_kernel(const float* __restrict__ in, const float* __restrict__ resid,
          const float* __restrict__ sc, const float* __restrict__ bi,
          float* __restrict__ out, int nrows) {
    int wave = threadIdx.x >> 5;
    int lane = threadIdx.x & 31;
    int row = blockIdx.x * 8 + wave;
    if (row >= nrows) return;
    const float4 v0 = *(const float4*)(in + (size_t)row * DD + lane * 4);
    float x0 = v0.x, x1 = v0.y, x2 = v0.z, x3 = v0.w;
    if (resid) {
        const float4 r0 = *(const float4*)(resid + (size_t)row * DD + lane * 4);
        x0 += r0.x; x1 += r0.y; x2 += r0.z; x3 += r0.w;
    }
    float s = x0 + x1 + x2 + x3;
    float q = x0 * x0 + x1 * x1 + x2 * x2 + x3 * x3;
    for (int off = 16; off > 0; off >>= 1) {
        s += __shfl_xor(s, off, 32);
        q += __shfl_xor(q, off, 32);
    }
    float mu = s * (1.0f / DD);
    float var = q * (1.0f / DD) - mu * mu;
    float inv = rsqrtf(var + LN_EPS);
    const float4 sv = *(const float4*)(sc + lane * 4);
    const float4 bv = *(const float4*)(bi + lane * 4);
    float4 o;
    o.x = (x0 - mu) * inv * sv.x + bv.x;
    o.y = (x1 - mu) * inv * sv.y + bv.y;
    o.z = (x2 - mu) * inv * sv.z + bv.z;
    o.w = (x3 - mu) * inv * sv.w + bv.w;
    *(float4*)(out + (size_t)row * DD + lane * 4) = o;
}

// ---------------- fp32 WMMA GEMM: C = act(A[M,K] @ W[K,NOUT] + bias) --------
// Compile-time K/NOUT so all B-matrix global loads and LDS A reads use
// immediate offsets (24-bit IOFFSET) instead of per-iteration 64-bit VALU
// address math. Block: 256 thr = 8 waves; covers 16 rows x 128 cols of C.
// Each wave: one 16x16 C tile via K/4 v_wmma_f32_16x16x4_f32 steps.
// LDS A panel: row stride K+4 floats -> 16B-aligned rows AND conflict-free
// (bank = (4*lane + k) mod 64, distinct over 16 lanes).
// Fragment layouts per ISA 7.12.2:
//   A 16x4:  lanes0-15 {K=k0,k0+1}, lanes16-31 {K=k0+2,k0+3}, M = lane&15
//   B 4x16:  vgpr0 = row (k0+2*half), vgpr1 = row (k0+2*half+1), N = lane&15
//   C/D:     vgpr j -> rows j (lanes0-15) and j+8 (lanes16-31), N = lane&15
template <int K, int NOUT, int ACT>
__global__ void __launch_bounds__(256)
gemm_wmma_f32(const float* __restrict__ A, const float* __restrict__ W,
              const float* __restrict__ bias, float* __restrict__ C,
              int M) {
    constexpr int LSTR = K + 4;
    __shared__ float As[16 * LSTR];
    int mbase = blockIdx.x * 16;
    if (mbase >= M) return;

    // cooperative A panel copy, vectorized float4 (global b128 -> ds b128)
    constexpr int NVEC = 16 * K / 4;
    for (int i = threadIdx.x; i < NVEC; i += 256) {
        int flat = i * 4;
        int r = flat / K;
        int kk = flat - r * K;
        float4 v = *(const float4*)(A + (size_t)(mbase + r) * K + kk);
        *(float4*)(&As[r * LSTR + kk]) = v;
    }
    __syncthreads();

    int wave = threadIdx.x >> 5;
    int lane = threadIdx.x & 31;
    int half = lane >> 4;
    int l16 = lane & 15;
    int nbase = (blockIdx.y * 8 + wave) * 16;
    if (nbase >= NOUT) return;

    const float* Asr = As + l16 * LSTR + 2 * half;
    const float* Wp = W + (size_t)(2 * half) * NOUT + nbase + l16;

    v8f acc = {};
#pragma unroll 8
    for (int k0 = 0; k0 < K; k0 += 4) {
        v2f a, b;
        a.x = Asr[k0];
        a.y = Asr[k0 + 1];
        b.x = Wp[k0 * NOUT];
        b.y = Wp[(k0 + 1) * NOUT];
        acc = __builtin_amdgcn_wmma_f32_16x16x4_f32(
            false, a, false, b, (short)0, acc, false, false);
    }

    int coln = nbase + l16;
    float bv = bias[coln];
#pragma unroll
    for (int j = 0; j < 8; ++j) {
        int mrow = mbase + j + 8 * half;
        float v = acc[j] + bv;
        if (ACT == 1) v = 0.5f * v * (1.0f + erff(v * 0.70710678118654752f));
        C[(size_t)mrow * NOUT + coln] = v;
    }
}

// ---------------- edge pass 1: attention logits + segment max ----------------
// thread t -> edge e = t>>3, head h = t&7
__global__ void __launch_bounds__(256)
edge_attn1(const float* __restrict__ Q, const float* __restrict__ Kv,
           const float* __restrict__ pos, const int* __restrict__ rowp,
           const int* __restrict__ colp, const float* __restrict__ ea,
           const float* __restrict__ We, const float* __restrict__ be,
           const float* __restrict__ Wg, const float* __restrict__ bg,
           float* __restrict__ ATT, float* __restrict__ MB, int nE) {
    int t = blockIdx.x * blockDim.x + threadIdx.x;
    int e = t >> 3;
    int h = t & 7;
    if (e >= nE) return;
    int r = rowp[e];
    int c = colp[e];
    float dx = pos[r * 3 + 0] - pos[c * 3 + 0];
    float dy = pos[r * 3 + 1] - pos[c * 3 + 1];
    float dz = pos[r * 3 + 2] - pos[c * 3 + 2];
    float dist = sqrtf(dx * dx + dy * dy + dz * dz);
    float g = 1.0f / (1.0f + expf(-(dist * Wg[h] + bg[h])));
    const float4* qr = (const float4*)(Q + (size_t)r * DD + h * DHH);
    const float4* kr = (const float4*)(Kv + (size_t)c * DD + h * DHH);
    float dot = 0.0f;
#pragma unroll
    for (int j = 0; j < 4; ++j) {
        float4 a = qr[j], b = kr[j];
        dot += a.x * b.x + a.y * b.y + a.z * b.z + a.w * b.w;
    }
    float bias = be[h];
    const float4* eap = (const float4*)(ea + (size_t)e * EDIM);
#pragma unroll
    for (int i = 0; i < 4; ++i) {
        float4 v = eap[i];
        bias += v.x * We[(i * 4 + 0) * HH + h] + v.y * We[(i * 4 + 1) * HH + h] +
                v.z * We[(i * 4 + 2) * HH + h] + v.w * We[(i * 4 + 3) * HH + h];
    }
    float attn = dot * 0.25f * g + bias;  // 1/sqrt(16) = 0.25
    ATT[t] = attn;
    atomicMaxF(&MB[r * HH + h], attn);
}

__global__ void __launch_bounds__(256)
fix_m_kernel(float* __restrict__ MB, int n) {
    int i = blockIdx.x * blockDim.x + threadIdx.x;
    if (i < n) {
        float m = MB[i];
        if (!isfinite(m)) MB[i] = 0.0f;
    }
}

// ---------------- edge pass 2: exp + segment sum ----------------
__global__ void __launch_bounds__(256)
edge_attn2(float* __restrict__ ATT, const float* __restrict__ MB,
           float* __restrict__ SB, const int* __restrict__ rowp, int nE) {
    int t = blockIdx.x * blockDim.x + threadIdx.x;
    int e = t >> 3;
    int h = t & 7;
    if (e >= nE) return;
    int r = rowp[e];
    float ex = expf(ATT[t] - MB[r * HH + h]);
    ATT[t] = ex;
    atomicAdd(&SB[r * HH + h], ex);
}

// ---------------- edge pass 3: weighted V scatter-add ----------------
// one wave32 per edge; lane owns 4 channels (one head per 4 lanes)
__global__ void __launch_bounds__(256)
edge_agg(const float* __restrict__ ATT, const float* __restrict__ SB,
         const float* __restrict__ V, const int* __restrict__ rowp,
         const int* __restrict__ colp, float* __restrict__ AGG, int nE) {
    int t = blockIdx.x * blockDim.x + threadIdx.x;
    int e = t >> 5;
    int lane = t & 31;
    if (e >= nE) return;
    int r = rowp[e];
    int c = colp[e];
    int h = lane >> 2;  // 4 channels per lane, 16 channels per head
    float a = ATT[(size_t)e * HH + h] / (SB[r * HH + h] + 1e-16f);
    const float4 vv = *(const float4*)(V + (size_t)c * DD + lane * 4);
    float* dst = AGG + (size_t)r * DD + lane * 4;
    atomicAdd(dst + 0, a * vv.x);
    atomicAdd(dst + 1, a * vv.y);
    atomicAdd(dst + 2, a * vv.z);
    atomicAdd(dst + 3, a * vv.w);
}

// ---------------- host-side orchestration ----------------
extern "C" void kernel_launch(void* const* d_in, const int* in_sizes, int n_in,
                              void* d_out, int out_size, void* d_ws, size_t ws_size,
                              hipStream_t stream) {
    (void)in_sizes; (void)n_in; (void)out_size; (void)ws_size;
    const float* x_in = (const float*)d_in[0];
    const float* pos  = (const float*)d_in[1];
    const int*   ei   = (const int*)d_in[2];   // [2,E] int32 (jax default non-x64)
    const float* ea   = (const float*)d_in[3];
    const int* rowp = ei;
    const int* colp = ei + EE;

    const int ND = NN * DD;
    float* X    = (float*)d_ws;    // current activations (post pre-norm)
    float* PROJ = X + ND;          // projection / FFN output
    float* AGG  = PROJ + ND;       // aggregated messages; H4 aliases AGG..V
    float* Q    = AGG + ND;
    float* Kb   = Q + ND;
    float* V    = Kb + ND;
    float* ATT  = V + ND;          // E*H
    float* MB   = ATT + (size_t)EE * HH;  // N*H segment max
    float* SB   = MB + NN * HH;           // N*H segment sum
    float* H4   = AGG;             // N*4D = 4*ND, exactly spans AGG,Q,Kb,V

    dim3 blk(256);
    dim3 gLN(NN / 8);                  // 6250
    dim3 gGemm128(NN / 16, 1);         // 3125 x 1  (NOUT=128)
    dim3 gGemm512(NN / 16, 4);         // 3125 x 4  (NOUT=512)
    dim3 gEH((EE * HH) / 256);         // 18750
    dim3 gEW((EE * 32) / 256);         // 75000
    dim3 gNH((NN * HH + 255) / 256);
    dim3 gND((ND + 255) / 256);

    for (int l = 0; l < NLAYER; ++l) {
        int pb = 4 + l * 22;
        const float* Wq = (const float*)d_in[pb + 0];
        const float* bq = (const float*)d_in[pb + 1];
        const float* Wk = (const float*)d_in[pb + 2];
        const float* bk = (const float*)d_in[pb + 3];
        const float* Wv = (const float*)d_in[pb + 4];
        const float* bv = (const float*)d_in[pb + 5];
        const float* We = (const float*)d_in[pb + 6];
        const float* be = (const float*)d_in[pb + 7];
        const float* Wg = (const float*)d_in[pb + 8];
        const float* bg = (const float*)d_in[pb + 9];
        const float* Wo = (const float*)d_in[pb + 10];
        const float* bo = (const float*)d_in[pb + 11];
        const float* W1 = (const float*)d_in[pb + 12];
        const float* b1 = (const float*)d_in[pb + 13];
        const float* W2 = (const float*)d_in[pb + 14];
        const float* b2 = (const float*)d_in[pb + 15];
        const float* pn_s = (const float*)d_in[pb + 16];
        const float* pn_b = (const float*)d_in[pb + 17];
        const float* n1_s = (const float*)d_in[pb + 18];
        const float* n1_b = (const float*)d_in[pb + 19];
        const float* n2_s = (const float*)d_in[pb + 20];
        const float* n2_b = (const float*)d_in[pb + 21];

        // pre-norm
        const float* xin = (l == 0) ? x_in : X;
        ln_kernel<<<gLN, blk, 0, stream>>>(xin, nullptr, pn_s, pn_b, X, NN);

        // QKV projections (WMMA fp32)
        gemm_wmma_f32<128, 128, 0><<<gGemm128, blk, 0, stream>>>(X, Wq, bq, Q, NN);
        gemm_wmma_f32<128, 128, 0><<<gGemm128, blk, 0, stream>>>(X, Wk, bk, Kb, NN);
        gemm_wmma_f32<128, 128, 0><<<gGemm128, blk, 0, stream>>>(X, Wv, bv, V, NN);

        // segment-softmax scratch init
        fill_kernel<<<gNH, blk, 0, stream>>>(MB, -INFINITY, NN * HH);
        fill_kernel<<<gNH, blk, 0, stream>>>(SB, 0.0f, NN * HH);
        fill_kernel<<<gND, blk, 0, stream>>>(AGG, 0.0f, ND);

        // edge attention
        edge_attn1<<<gEH, blk, 0, stream>>>(Q, Kb, pos, rowp, colp, ea, We, be,
                                            Wg, bg, ATT, MB, EE);
        fix_m_kernel<<<gNH, blk, 0, stream>>>(MB, NN * HH);
        edge_attn2<<<gEH, blk, 0, stream>>>(ATT, MB, SB, rowp, EE);
        edge_agg<<<gEW, blk, 0, stream>>>(ATT, SB, V, rowp, colp, AGG, EE);

        // output projection + residual LN
        gemm_wmma_f32<128, 128, 0><<<gGemm128, blk, 0, stream>>>(AGG, Wo, bo, PROJ, NN);
        ln_kernel<<<gLN, blk, 0, stream>>>(X, PROJ, n1_s, n1_b, X, NN);

        // FFN: gelu(X@W1+b1)@W2+b2, residual LN
        gemm_wmma_f32<128, 512, 1><<<gGemm512, blk, 0, stream>>>(X, W1, b1, H4, NN);
        gemm_wmma_f32<512, 128, 0><<<gGemm128, blk, 0, stream>>>(H4, W2, b2, PROJ, NN);
        float* xout = (l == NLAYER - 1) ? (float*)d_out : X;
        ln_kernel<<<gLN, blk, 0, stream>>>(X, PROJ, n2_s, n2_b, xout, NN);
    }
}